// CustomModel_7378753814838
// MI455X (gfx1250) — compile-verified
//
#include <hip/hip_runtime.h>

typedef __attribute__((ext_vector_type(2))) float v2f;
typedef __attribute__((ext_vector_type(8))) float v8f;

#define RDIM 128
#define FDIM 128
#define KK 4
#define LDSTR 132   // padded LDS row stride (floats): +4 keeps ds_load_b64 conflict-free

__global__ __launch_bounds__(256)
void fused_rbf_softmax_kernel(const float* __restrict__ x1,
                              const float* __restrict__ x2,
                              const float* __restrict__ sigmas,
                              const float* __restrict__ means,
                              const float* __restrict__ sigma_params,
                              float* __restrict__ out)
{
    __shared__ float sA[RDIM * LDSTR];   // A tile; later reused to hold dot[i][j]
    __shared__ float sB[RDIM * LDSTR];   // B tile
    __shared__ float sSa[RDIM], sSqa[RDIM], sSb[RDIM], sSqb[RDIM];
    __shared__ float sRcp[KK * RDIM];    // 1 / sum_j exp(kernel_value)

    const int t    = threadIdx.x;
    const int n    = blockIdx.x;
    const int lane = t & 31;
    const int wave = t >> 5;

    // ---------------- phase 0: stage A(n), B(n) into LDS (float4, 16B aligned) ----
    const float* a = x1 + (size_t)n * (RDIM * FDIM);
    const float* b = x2 + (size_t)n * (RDIM * FDIM);
    #pragma unroll
    for (int r = 0; r < 16; ++r) {
        int e = t + 256 * r;           // float4 index 0..4095
        int i = e >> 5;                // row
        int f = (e & 31) << 2;         // col (multiple of 4)
        float4 va = ((const float4*)a)[e];
        float4 vb = ((const float4*)b)[e];
        *(float4*)&sA[i * LDSTR + f] = va;
        *(float4*)&sB[i * LDSTR + f] = vb;
    }
    __syncthreads();

    // ---------------- phase 1: row aggregates (sum, sumsq) -----------------------
    {
        int row = t & 127;
        const float* p = ((t < 128) ? sA : sB) + row * LDSTR;
        float s = 0.f, sq = 0.f;
        for (int f = 0; f < FDIM; ++f) { float v = p[f]; s += v; sq += v * v; }
        if (t < 128) { sSa[row] = s; sSqa[row] = sq; }
        else         { sSb[row] = s; sSqb[row] = sq; }
    }

    // ---------------- phase 2: WMMA fp32 GEMM  dot = A * B^T ---------------------
    // wave w owns i-tile w (rows 16w..16w+15), all 8 j-tiles.
    v8f acc[8] = {};
    const int iRow  = 16 * wave + (lane & 15);
    const int fHalf = (lane >> 4) << 1;        // lanes 16-31 hold K+2,K+3
    for (int ks = 0; ks < 32; ++ks) {
        int f0 = 4 * ks + fHalf;
        v2f afrag = *(const v2f*)&sA[iRow * LDSTR + f0];
        #pragma unroll
        for (int jt = 0; jt < 8; ++jt) {
            int jRow = 16 * jt + (lane & 15);
            v2f bfrag = *(const v2f*)&sB[jRow * LDSTR + f0];
            acc[jt] = __builtin_amdgcn_wmma_f32_16x16x4_f32(
                false, afrag, false, bfrag, (short)0, acc[jt], false, false);
        }
    }
    __syncthreads();   // all waves done reading sA before we overwrite it

    // ---------------- phase 3: spill dot tiles into LDS (over sA) ----------------
    {
        int iBase = 16 * wave + 8 * (lane >> 4);   // C layout: VGPR r -> M = r (+8 for hi half)
        int j0 = lane & 15;
        #pragma unroll
        for (int jt = 0; jt < 8; ++jt)
            #pragma unroll
            for (int r = 0; r < 8; ++r)
                sA[(iBase + r) * LDSTR + 16 * jt + j0] = acc[jt][r];
    }
    __syncthreads();

    // ---------------- per-k parameters (uniform, tiny) ---------------------------
    float mk_[KK], inv2s[KK], wgt[KK];
    {
        float rk[KK], wm = -3.0e38f;
        #pragma unroll
        for (int k = 0; k < KK; ++k) {
            mk_[k] = means[k];
            float sg = sigmas[k];
            inv2s[k] = 1.0f / (2.0f * sg * sg);
            float sp = sigma_params[k];
            rk[k] = 1.0f / (sp * sp);
            wm = fmaxf(wm, rk[k]);
        }
        float ws = 0.f;
        #pragma unroll
        for (int k = 0; k < KK; ++k) { wgt[k] = __expf(rk[k] - wm); ws += wgt[k]; }
        float inv = 1.0f / ws;
        #pragma unroll
        for (int k = 0; k < KK; ++k) wgt[k] *= inv;
    }

    // ---------------- phase 4: per (k,i) softmax denominator ---------------------
    // kernel_value = exp(-d2/(2s^2)) in (0,1]  (d2 = ||a_i - b_j - m||^2 >= 0),
    // so softmax over j needs no max-subtraction: S = sum_j exp(kv).
    for (int task = t; task < KK * RDIM; task += 256) {
        int k = task >> 7, i = task & 127;
        float mk = mk_[k], is = inv2s[k];
        float c0 = sSqa[i] - 2.0f * mk * sSa[i] + (float)FDIM * mk * mk;
        float s = 0.f;
        for (int j = 0; j < RDIM; ++j) {
            float d  = sA[i * LDSTR + j];
            float d2 = c0 + sSqb[j] + 2.0f * mk * sSb[j] - 2.0f * d;
            float kv = __expf(-d2 * is);
            s += __expf(kv);
        }
        sRcp[(k << 7) + i] = 1.0f / s;
    }
    __syncthreads();

    // ---------------- phase 5: weighted combine, coalesced store -----------------
    float* o = out + (size_t)n * (RDIM * RDIM);
    for (int e = 0; e < 64; ++e) {
        int idx = t + 256 * e;
        int i = idx >> 7, j = idx & 127;
        float d    = sA[i * LDSTR + j];
        float base = sSqb[j] - 2.0f * d;
        float sai = sSa[i], sqai = sSqa[i], sbj = sSb[j];
        float r = 0.f;
        #pragma unroll
        for (int k = 0; k < KK; ++k) {
            float mk = mk_[k];
            float d2 = (sqai - 2.0f * mk * sai + (float)FDIM * mk * mk)
                     + base + 2.0f * mk * sbj;
            float kv = __expf(-d2 * inv2s[k]);
            r += wgt[k] * __expf(kv) * sRcp[(k << 7) + i];
        }
        o[idx] = r;
    }
}

extern "C" void kernel_launch(void* const* d_in, const int* in_sizes, int n_in,
                              void* d_out, int out_size, void* d_ws, size_t ws_size,
                              hipStream_t stream) {
    const float* x1 = (const float*)d_in[0];
    const float* x2 = (const float*)d_in[1];
    const float* sigmas = (const float*)d_in[2];
    const float* means = (const float*)d_in[3];
    const float* sigma_params = (const float*)d_in[4];
    float* out = (float*)d_out;

    int N = in_sizes[0] / (RDIM * FDIM);   // 128
    fused_rbf_softmax_kernel<<<N, 256, 0, stream>>>(x1, x2, sigmas, means,
                                                    sigma_params, out);
}